// RetrievalModel_90598040141832
// MI455X (gfx1250) — compile-verified
//
#include <hip/hip_runtime.h>
#include <math.h>

typedef __attribute__((ext_vector_type(2))) float v2f;
typedef __attribute__((ext_vector_type(8))) float v8f;

#define B_N    4096
#define D_IN   768
#define D_F    256

// ---------------------------------------------------------------------------
// Kernel 1: feature projection  F = X(4096x768) @ W(768x256) + bias
// One wave32 computes one 16x16 output tile with V_WMMA_F32_16X16X4_F32,
// looping K in steps of 4. 8 waves per block.
// A-frag layout (16x4 f32): lane L<16 -> row L, cols {k,k+1};
//                           lane L>=16 -> row L-16, cols {k+2,k+3}.
// B-frag layout (4x16 f32): lane L<16 -> col L, rows {k,k+1};
//                           lane L>=16 -> col L-16, rows {k+2,k+3}.
// C/D (16x16 f32): vgpr v, lanes 0-15 -> (M=v, N=lane); lanes 16-31 -> (M=v+8).
// ---------------------------------------------------------------------------
__global__ __launch_bounds__(256) void proj_wmma_kernel(
    const float* __restrict__ X, const float* __restrict__ W,
    const float* __restrict__ bias, float* __restrict__ F)
{
    const int lane = threadIdx.x & 31;
    const int wave = threadIdx.x >> 5;
    const int tile = blockIdx.x * 8 + wave;
    const int ntN  = D_F / 16;              // 16 N-tiles
    const int ti   = tile / ntN;
    const int tj   = tile % ntN;
    const int i0   = ti * 16;
    const int j0   = tj * 16;
    const int lr   = lane & 15;
    const int hi   = lane >> 4;             // 0 or 1

    const float* arow = X + (size_t)(i0 + lr) * D_IN;

    v8f acc = {};
    for (int kk = 0; kk < D_IN; kk += 4) {
        const int ka = kk + 2 * hi;
        v2f a = *(const v2f*)(arow + ka);
        v2f b;
        b.x = W[(size_t)(ka    ) * D_F + (j0 + lr)];
        b.y = W[(size_t)(ka + 1) * D_F + (j0 + lr)];
        acc = __builtin_amdgcn_wmma_f32_16x16x4_f32(
                  false, a, false, b, (short)0, acc, false, false);
    }

    const float bv = bias[j0 + lr];
#pragma unroll
    for (int v = 0; v < 8; ++v) {
        const int r = i0 + v + 8 * hi;
        F[(size_t)r * D_F + (j0 + lr)] = acc[v] + bv;
    }
}

// ---------------------------------------------------------------------------
// Kernel 2: per-row stats. One wave32 per row (lane l covers 8 floats).
//   norm_img[i] = ||F_img[i]||,  norm_txt[i] = ||F_txt[i]||
//   s_img[i] = F_img[i].w1 + cs_b,  s_txt[i] = F_txt[i].w2
// ---------------------------------------------------------------------------
__global__ __launch_bounds__(256) void rowstats_kernel(
    const float* __restrict__ F_img, const float* __restrict__ F_txt,
    const float* __restrict__ cs_w,  const float* __restrict__ cs_b,
    float* __restrict__ norm_img, float* __restrict__ norm_txt,
    float* __restrict__ s_img,    float* __restrict__ s_txt)
{
    const int lane = threadIdx.x & 31;
    const int wave = threadIdx.x >> 5;
    const int row  = blockIdx.x * 8 + wave;

    const float* fi = F_img + (size_t)row * D_F + lane * 8;
    const float* ft = F_txt + (size_t)row * D_F + lane * 8;
    const float* w1 = cs_w + lane * 8;
    const float* w2 = cs_w + D_F + lane * 8;

    float ssi = 0.f, sst = 0.f, di = 0.f, dt = 0.f;
#pragma unroll
    for (int q = 0; q < 8; ++q) {
        const float a = fi[q], b = ft[q];
        ssi += a * a;       sst += b * b;
        di  += a * w1[q];   dt  += b * w2[q];
    }
    // wave32 tree reduction
#pragma unroll
    for (int m = 16; m >= 1; m >>= 1) {
        ssi += __shfl_xor(ssi, m, 32);
        sst += __shfl_xor(sst, m, 32);
        di  += __shfl_xor(di,  m, 32);
        dt  += __shfl_xor(dt,  m, 32);
    }
    if (lane == 0) {
        norm_img[row] = sqrtf(ssi);
        norm_txt[row] = sqrtf(sst);
        s_img[row]    = di + cs_b[0];   // fold bias into image term
        s_txt[row]    = dt;
    }
}

// ---------------------------------------------------------------------------
// Kernel 3: sim = F_img @ F_txt^T, fused epilogue producing all 3 outputs.
// One wave32 per 16x16 tile; K=256 in steps of 4 via V_WMMA_F32_16X16X4_F32.
// Both operands are contiguous float2 reads from the (L2-resident) features.
// ---------------------------------------------------------------------------
__global__ __launch_bounds__(256) void sim_wmma_kernel(
    const float* __restrict__ F_img, const float* __restrict__ F_txt,
    const float* __restrict__ norm_img, const float* __restrict__ norm_txt,
    const float* __restrict__ s_img,    const float* __restrict__ s_txt,
    const float* __restrict__ edge,
    float* __restrict__ out_final, float* __restrict__ out_sim,
    float* __restrict__ out_causal)
{
    const int lane = threadIdx.x & 31;
    const int wave = threadIdx.x >> 5;
    const int tile = blockIdx.x * 8 + wave;
    const int ntN  = B_N / 16;              // 256 N-tiles
    const int i0   = (tile / ntN) * 16;
    const int j0   = (tile % ntN) * 16;
    const int lr   = lane & 15;
    const int hi   = lane >> 4;

    const float* arow = F_img + (size_t)(i0 + lr) * D_F + 2 * hi;
    const float* brow = F_txt + (size_t)(j0 + lr) * D_F + 2 * hi;

    v8f acc = {};
#pragma unroll 4
    for (int kk = 0; kk < D_F; kk += 4) {
        v2f a = *(const v2f*)(arow + kk);
        v2f b = *(const v2f*)(brow + kk);
        acc = __builtin_amdgcn_wmma_f32_16x16x4_f32(
                  false, a, false, b, (short)0, acc, false, false);
    }

    const float rn_t = norm_txt[j0 + lr];
    const float st_t = s_txt[j0 + lr];
#pragma unroll
    for (int v = 0; v < 8; ++v) {
        const int r   = i0 + v + 8 * hi;
        const size_t idx = (size_t)r * B_N + (j0 + lr);
        const float simv     = acc[v] / (norm_img[r] * rn_t);
        const float strength = s_img[r] + st_t;          // cs_b already folded
        const float causalv  = edge[idx] * strength;
        out_final[idx]  = simv + 0.3f * causalv;
        out_sim[idx]    = simv;
        out_causal[idx] = causalv;
    }
}

// ---------------------------------------------------------------------------
extern "C" void kernel_launch(void* const* d_in, const int* in_sizes, int n_in,
                              void* d_out, int out_size, void* d_ws, size_t ws_size,
                              hipStream_t stream)
{
    const float* images = (const float*)d_in[0];
    const float* texts  = (const float*)d_in[1];
    const float* W_img  = (const float*)d_in[2];
    const float* b_img  = (const float*)d_in[3];
    const float* W_txt  = (const float*)d_in[4];
    const float* b_txt  = (const float*)d_in[5];
    const float* cs_w   = (const float*)d_in[6];
    const float* cs_b   = (const float*)d_in[7];
    const float* edge   = (const float*)d_in[8];

    // Workspace layout (floats): F_img | F_txt | norm_img | norm_txt | s_img | s_txt
    float* F_img    = (float*)d_ws;
    float* F_txt    = F_img + (size_t)B_N * D_F;
    float* norm_img = F_txt + (size_t)B_N * D_F;
    float* norm_txt = norm_img + B_N;
    float* s_img    = norm_txt + B_N;
    float* s_txt    = s_img + B_N;

    float* out_final  = (float*)d_out;
    float* out_sim    = out_final + (size_t)B_N * B_N;
    float* out_causal = out_sim + (size_t)B_N * B_N;

    // Stage 1: projections. tiles = (4096/16)*(256/16) = 4096; 8 waves/block.
    const int proj_blocks = (B_N / 16) * (D_F / 16) / 8;   // 512
    proj_wmma_kernel<<<proj_blocks, 256, 0, stream>>>(images, W_img, b_img, F_img);
    proj_wmma_kernel<<<proj_blocks, 256, 0, stream>>>(texts,  W_txt, b_txt, F_txt);

    // Stage 2: row stats. 4096 rows, 8 rows (waves) per block.
    rowstats_kernel<<<B_N / 8, 256, 0, stream>>>(F_img, F_txt, cs_w, cs_b,
                                                 norm_img, norm_txt, s_img, s_txt);

    // Stage 3: fused sim/causal/final. tiles = (4096/16)^2 = 65536; 8/block.
    const int sim_blocks = (B_N / 16) * (B_N / 16) / 8;    // 8192
    sim_wmma_kernel<<<sim_blocks, 256, 0, stream>>>(F_img, F_txt,
                                                    norm_img, norm_txt, s_img, s_txt,
                                                    edge, out_final, out_sim, out_causal);
}